// SparseAttention_33809982554281
// MI455X (gfx1250) — compile-verified
//
#include <hip/hip_runtime.h>

typedef __attribute__((ext_vector_type(2))) float v2f;
typedef __attribute__((ext_vector_type(8))) float v8f;

#ifndef __has_builtin
#define __has_builtin(x) 0
#endif

#if __has_builtin(__builtin_amdgcn_wmma_f32_16x16x4_f32)
#define HAVE_WMMA_F32X4 1
#else
#define HAVE_WMMA_F32X4 0
#endif

// ---------------------------------------------------------------------------
// Stage 1: logits[b,t] = sum_d x[b,t,d] * W[d]
// One wave per 16-row (t) tile. WMMA f32 16x16x4: A = x tile rows (16 x 4),
// B = W[k] replicated across all 16 columns -> every column of D holds the
// 16 logits. 128 WMMAs per tile accumulate the K=512 reduction in the matrix
// pipe. W staged in LDS once per block.
// ---------------------------------------------------------------------------
__launch_bounds__(256)
__global__ void sa_logits_kernel(const float* __restrict__ x,
                                 const float* __restrict__ W,
                                 float* __restrict__ logits,
                                 int T) {
  constexpr int D = 512;
  __shared__ float Ws[D];
  for (int i = threadIdx.x; i < D; i += 256) Ws[i] = W[i];
  __syncthreads();

  const int wave = threadIdx.x >> 5;
  const int lane = threadIdx.x & 31;
  const long tile = (long)blockIdx.x * 8 + wave;   // 16 t-rows per tile
  const int tilesPerB = T >> 4;
  const int b  = (int)(tile / tilesPerB);
  const int t0 = (int)(tile % tilesPerB) << 4;

  // A-matrix 16x4 f32 layout: lanes 0-15 -> M=lane, VGPR0=K0,VGPR1=K1;
  // lanes 16-31 -> M=lane-16, VGPR0=K2,VGPR1=K3.
  const int m     = lane & 15;
  const int khalf = (lane >> 4) << 1;              // 0 or 2
  const float* __restrict__ xr =
      x + ((long)b * T + t0) * D + (long)m * D + khalf;

#if HAVE_WMMA_F32X4
  v8f c = {};
  for (int k0 = 0; k0 < D; k0 += 4) {
    if ((k0 & 63) == 0) __builtin_prefetch(xr + k0 + 256, 0, 0);
    v2f a, bf;
    a.x  = xr[k0];
    a.y  = xr[k0 + 1];
    bf.x = Ws[k0 + khalf];        // B[k][n] = W[k] for all n (broadcast cols)
    bf.y = Ws[k0 + khalf + 1];
    c = __builtin_amdgcn_wmma_f32_16x16x4_f32(false, a, false, bf,
                                              (short)0, c, false, false);
  }
  // D layout: VGPR r -> row r (lanes 0-15) / row 8+r (lanes 16-31); all
  // columns identical, so lane 0 carries rows 0-7, lane 16 carries rows 8-15.
  if (lane == 0 || lane == 16) {
    float* lp = logits + (long)b * T + t0 + ((lane >> 4) << 3);
#pragma unroll
    for (int r = 0; r < 8; ++r) lp[r] = c[r];
  }
#else
  // VALU fallback: lanes m and m+16 each cover k%4 in {0,1} / {2,3}.
  float acc = 0.f;
  for (int k0 = 0; k0 < D; k0 += 4) {
    acc = fmaf(xr[k0],     Ws[k0 + khalf],     acc);
    acc = fmaf(xr[k0 + 1], Ws[k0 + khalf + 1], acc);
  }
  acc += __shfl_xor(acc, 16, 32);
  if (lane < 16) logits[(long)b * T + t0 + m] = acc;
#endif
}

// ---------------------------------------------------------------------------
// Stage 2: in-place sparsemax per row via Michelot fixed-point iteration.
// tau_{n+1} = (sum_{z>tau_n} z - 1) / |{z>tau_n}| ; tau is monotone
// non-decreasing so the active set shrinks; equal counts => converged (exact).
// One 256-thread block per batch row; row cached in LDS.
// ---------------------------------------------------------------------------
__launch_bounds__(256)
__global__ void sa_sparsemax_kernel(float* __restrict__ z, int T) {
  __shared__ float zr[4096];
  __shared__ float rs[256];
  __shared__ int   rc[256];
  const int tid = threadIdx.x;
  float* __restrict__ zrow = z + (long)blockIdx.x * T;

  for (int i = tid; i < T; i += 256) zr[i] = zrow[i];
  __syncthreads();

  // tau_0 with full support
  float ls = 0.f;
  for (int i = tid; i < T; i += 256) ls += zr[i];
  rs[tid] = ls;
  __syncthreads();
  for (int off = 128; off > 0; off >>= 1) {
    if (tid < off) rs[tid] += rs[tid + off];
    __syncthreads();
  }
  float tau = (rs[0] - 1.f) / (float)T;
  __syncthreads();

  int prev = T;
  for (int it = 0; it < 64; ++it) {
    float s = 0.f; int c = 0;
    for (int i = tid; i < T; i += 256) {
      float v = zr[i];
      if (v > tau) { s += v; ++c; }
    }
    rs[tid] = s; rc[tid] = c;
    __syncthreads();
    for (int off = 128; off > 0; off >>= 1) {
      if (tid < off) { rs[tid] += rs[tid + off]; rc[tid] += rc[tid + off]; }
      __syncthreads();
    }
    s = rs[0]; c = rc[0];
    __syncthreads();
    if (c == prev || c == 0) break;   // active set unchanged -> tau exact
    tau = (s - 1.f) / (float)c;
    prev = c;
  }

  for (int i = tid; i < T; i += 256) zrow[i] = fmaxf(zr[i] - tau, 0.f);
}

// ---------------------------------------------------------------------------
// Stage 3: out[b,d] = sum_t att[b,t] * x[b,t,d].
// Blocks own 128-t chunks; att chunk staged in LDS; x rows with att==0 are
// skipped entirely (uniform branch -> sparsemax sparsity kills pass-2 HBM
// traffic). Partials combined with global f32 atomics into zero-initialized
// output.
// ---------------------------------------------------------------------------
__launch_bounds__(512)
__global__ void sa_pool_kernel(const float* __restrict__ x,
                               const float* __restrict__ att,
                               float* __restrict__ out,
                               int T) {
  constexpr int CH = 128;
  const int nchunk = T / CH;
  const int b  = blockIdx.x / nchunk;
  const int t0 = (blockIdx.x % nchunk) * CH;
  __shared__ float as[CH];
  const int tid = threadIdx.x;               // d index, 512 threads
  if (tid < CH) as[tid] = att[(long)b * T + t0 + tid];
  __syncthreads();

  const float* __restrict__ xp = x + ((long)b * T + t0) * 512 + tid;
  float acc = 0.f;
  for (int t = 0; t < CH; ++t) {
    float a = as[t];
    if (a != 0.f) acc = fmaf(a, xp[(long)t * 512], acc);
  }
  if (acc != 0.f) atomicAdd(&out[b * 512 + tid], acc);
}

// ---------------------------------------------------------------------------
extern "C" void kernel_launch(void* const* d_in, const int* in_sizes, int n_in,
                              void* d_out, int out_size, void* d_ws, size_t ws_size,
                              hipStream_t stream) {
  (void)n_in; (void)out_size; (void)ws_size;
  const float* x  = (const float*)d_in[0];
  const float* W  = (const float*)d_in[1];
  float* out = (float*)d_out;
  float* att = (float*)d_ws;                 // B*T floats of scratch

  const int T = 4096, D = 512;
  const int B = in_sizes[0] / (T * D);

  // Stage 1: 8 waves/block, 16 t-rows per wave -> 128 t per block
  sa_logits_kernel<<<dim3((B * T) / 128), 256, 0, stream>>>(x, W, att, T);
  // Stage 2: one block per batch row, in-place logits -> att
  sa_sparsemax_kernel<<<dim3(B), 256, 0, stream>>>(att, T);
  // Stage 3: zero output, then sparse-aware pooling with f32 atomics
  hipMemsetAsync(d_out, 0, (size_t)B * D * sizeof(float), stream);
  sa_pool_kernel<<<dim3(B * (T / 128)), 512, 0, stream>>>(x, att, out, T);
}